// GAENodeClassification_28767690948710
// MI455X (gfx1250) — compile-verified
//
#include <hip/hip_runtime.h>
#include <hip/hip_bf16.h>

// ---------------------------------------------------------------------------
// GCN node classification on gfx1250 (CDNA5), fp32 WMMA path.
//   h0 = emb[x];  h1 = relu(gcnconv(h0, W1, b1));  z = gcnconv(h1, W2, b2)
//   out = mean(z[x_position]) @ Wc + bc          (mean/@ commute with +bc)
//
// CDNA5 specifics used:
//   * V_WMMA_F32_16X16X4_F32 (exact fp32 matmul; compute is tiny vs the
//     memory-bound edge scatter, so no reason to drop precision)
//   * global_load_async_to_lds_b128 + s_wait_asynccnt (inline asm) to stage
//     the block-shared 16xK A tile into LDS once (8x load-traffic reduction)
//   * ds_load_b64 fragment reads with K+4 LDS row padding (bank-conflict free
//     for the 32-lane x 2-word fragment access pattern)
//   * native global_atomic_add_f32 scatter via unsafeAtomicAdd
// ---------------------------------------------------------------------------

#define N_NODES 50000
#define N_EDGES 800000
#define VOCAB   4096
#define EMB_DIM 128
#define H2      256   // 2*HIDDEN
#define H1      128   // HIDDEN
#define N_POS   1024

typedef __attribute__((ext_vector_type(2))) float v2f;
typedef __attribute__((ext_vector_type(8))) float v8f;

__device__ __forceinline__ void atomAddF(float* p, float v) {
    unsafeAtomicAdd(p, v);   // native global_atomic_add_f32 (no CAS loop)
}

// gfx1250 async global->LDS copy, 16 bytes per lane, tracked by ASYNCcnt.
// dsaddr VGPR takes the LDS byte offset (low 32 bits of the flat shared addr).
__device__ __forceinline__ void async_g2l_b128(void* ldsPtr, const void* gPtr) {
    unsigned l = (unsigned)(uintptr_t)ldsPtr;
    asm volatile("global_load_async_to_lds_b128 %0, %1, off"
                 :: "v"(l), "v"(gPtr) : "memory");
}
__device__ __forceinline__ void wait_async0() {
    asm volatile("s_wait_asynccnt 0x0" ::: "memory");
}

// ---- degree / mask init ----------------------------------------------------
__global__ void k_init(float* dinv, int* mask) {
    int i = blockIdx.x * blockDim.x + threadIdx.x;
    if (i < N_NODES) { dinv[i] = 1.0f; mask[i] = 0; }   // 1.0 = self loop
}

__global__ void k_mask(const int* __restrict__ pos, int* mask) {
    int p = blockIdx.x * blockDim.x + threadIdx.x;
    if (p < N_POS) mask[pos[p]] = 1;
}

__global__ void k_deg(const int* __restrict__ col, float* deg) {
    int e = blockIdx.x * blockDim.x + threadIdx.x;
    if (e < N_EDGES) atomAddF(&deg[col[e]], 1.0f);
}

__global__ void k_rsqrt(float* dinv) {
    int i = blockIdx.x * blockDim.x + threadIdx.x;
    if (i < N_NODES) dinv[i] = rsqrtf(dinv[i]);         // deg >= 1 always
}

// ---- fp32 WMMA GEMM: C[M,N] = op(A)[M,K] @ B[K,N] --------------------------
// Block = 256 threads = 8 waves; block tile = 16 rows x min(N,128) cols,
// one 16x16 C tile per wave. A tile is async-staged into LDS once per block.
// A row r is Abase[gidx[r]] when GATHER (embedding lookup); RELU fuses the
// layer-1 activation into the layer-2 A-fragment read.
// Requires M%16==0, N%16==0, K%4==0 (all hold here).
template<int GATHER, int RELU, int K, int N>
__global__ void k_gemm_wmma(const float* __restrict__ Abase,
                            const int*   __restrict__ gidx,
                            const float* __restrict__ B,
                            float*       __restrict__ C) {
    constexpr int STRIDEA = K + 4;              // pad -> conflict-free ds_load_b64
    __shared__ float tileA[16 * STRIDEA];

    const int tid  = threadIdx.x;
    const int lane = tid & 31;
    const int wave = tid >> 5;
    const int rowBase = blockIdx.x << 4;

    // ---- stage A tile (16 x K fp32) into LDS via async b128 copies ----
    constexpr int CHUNKS_PER_ROW = K >> 2;      // 16B chunks per row
    constexpr int TOTAL_CHUNKS   = 16 * CHUNKS_PER_ROW;
    #pragma unroll
    for (int ch = 0; ch < TOTAL_CHUNKS; ch += 256) {
        const int c  = ch + tid;
        const int rr = c / CHUNKS_PER_ROW;
        const int cc = c - rr * CHUNKS_PER_ROW;
        int arow = rowBase + rr;
        if (GATHER) arow = gidx[arow];
        async_g2l_b128(&tileA[rr * STRIDEA + (cc << 2)],
                       Abase + (size_t)arow * K + (cc << 2));
    }
    wait_async0();          // own async copies done (LDS writes landed)
    __syncthreads();        // make all waves' copies visible to all waves

    // ---- WMMA K-loop ----
    const int m     = lane & 15;                // tile row / B-N lane id
    const int khalf = (lane >> 4) << 1;         // lanes 16-31 hold K+2,K+3
    const float* Al = &tileA[m * STRIDEA + khalf];
    const int n0 = (blockIdx.y << 7) + (wave << 4);
    const int n  = n0 + m;

    v8f c = {};
    #pragma unroll 4
    for (int k = 0; k < K; k += 4) {
        v2f a = *(const v2f*)(Al + k);          // ds_load_b64 fragment
        if (RELU) { a.x = fmaxf(a.x, 0.0f); a.y = fmaxf(a.y, 0.0f); }
        const int krow = k + khalf;
        v2f b;
        b.x = B[(size_t)krow * N + n];
        b.y = B[(size_t)(krow + 1) * N + n];
        c = __builtin_amdgcn_wmma_f32_16x16x4_f32(
                /*neg_a=*/false, a, /*neg_b=*/false, b,
                /*c_mod=*/(short)0, c, /*reuse_a=*/false, /*reuse_b=*/false);
    }

    // C/D layout: VGPR rr -> M=rr (lanes 0-15), M=rr+8 (lanes 16-31); N=lane&15
    float* Cp = C + (size_t)(rowBase + ((lane >> 4) << 3)) * N + n;
    #pragma unroll
    for (int rr = 0; rr < 8; ++rr) Cp[(size_t)rr * N] = c[rr];
}

// ---- self-loop term + bias: dst[i,j] = src[i,j]*dinv[i]^2 + bias[j] --------
__global__ void k_self_bias(const float* __restrict__ src,
                            float*       __restrict__ dst,
                            const float* __restrict__ dinv,
                            const float* __restrict__ bias,
                            unsigned total, int dimShift) {
    unsigned t = blockIdx.x * blockDim.x + threadIdx.x;
    if (t >= total) return;
    int i = (int)(t >> dimShift);
    int j = (int)(t & ((1u << dimShift) - 1u));
    float d = dinv[i];
    dst[t] = src[t] * d * d + bias[j];
}

// ---- edge-parallel normalized scatter-add ----------------------------------
// dst[col[e], :] += src[row[e], :] * dinv[row]*dinv[col]; optional dest mask.
__global__ void k_edge_scatter(const float* __restrict__ src,
                               float*       __restrict__ dst,
                               const float* __restrict__ dinv,
                               const int*   __restrict__ row,
                               const int*   __restrict__ col,
                               const int*   __restrict__ mask,
                               int dimShift) {
    unsigned t = blockIdx.x * blockDim.x + threadIdx.x;
    int e = (int)(t >> dimShift);
    if (e >= N_EDGES) return;
    int j = (int)(t & ((1u << dimShift) - 1u));
    int rs = row[e], cd = col[e];
    if (mask && !mask[cd]) return;
    float nrm = dinv[rs] * dinv[cd];
    atomAddF(dst + (((size_t)cd << dimShift) + j),
             src[((size_t)rs << dimShift) + j] * nrm);
}

// ---- mean of z rows at the 1024 positions ----------------------------------
__global__ void k_mean_pos(const float* __restrict__ z,
                           const int*   __restrict__ pos,
                           float*       __restrict__ meanz) {
    int j = threadIdx.x;                       // 128 threads
    float s = 0.0f;
    for (int p = 0; p < N_POS; ++p)
        s += z[(size_t)pos[p] * H1 + j];
    meanz[j] = s * (1.0f / (float)N_POS);
}

// ---- final matvec: out[c] = bc[c] + sum_k meanz[k] * Wc[k,c] ---------------
__global__ void k_matvec(const float* __restrict__ meanz,
                         const float* __restrict__ Wc,
                         const float* __restrict__ bc,
                         float*       __restrict__ out) {
    __shared__ float mlds[H1];
    if (threadIdx.x < H1) mlds[threadIdx.x] = meanz[threadIdx.x];
    __syncthreads();
    int c = blockIdx.x * blockDim.x + threadIdx.x;   // 4096 outputs
    float s = bc[c];
    #pragma unroll 4
    for (int k = 0; k < H1; ++k)
        s += mlds[k] * Wc[(size_t)k * VOCAB + c];
    out[c] = s;
}

// ---------------------------------------------------------------------------
extern "C" void kernel_launch(void* const* d_in, const int* in_sizes, int n_in,
                              void* d_out, int out_size, void* d_ws, size_t ws_size,
                              hipStream_t stream) {
    const int*   x    = (const int*)  d_in[0];           // (50000,1)
    const int*   ei   = (const int*)  d_in[1];           // (2, 800000)
    const int*   xpos = (const int*)  d_in[2];           // (1024,)
    const float* emb  = (const float*)d_in[3];           // (4096,128)
    const float* W1   = (const float*)d_in[4];           // (128,256)
    const float* b1   = (const float*)d_in[5];           // (256,)
    const float* W2   = (const float*)d_in[6];           // (256,128)
    const float* b2   = (const float*)d_in[7];           // (128,)
    const float* Wc   = (const float*)d_in[8];           // (128,4096)
    const float* bc   = (const float*)d_in[9];           // (4096,)
    float* out = (float*)d_out;                          // (1,4096)

    const int* erow = ei;
    const int* ecol = ei + N_EDGES;

    // workspace carve-up (floats)
    float* ws    = (float*)d_ws;
    float* dinv  = ws;                                   // 50000
    int*   mask  = (int*)(ws + N_NODES);                 // 50000
    float* hW1   = ws + 2 * N_NODES;                     // 50000*256
    float* out1  = hW1  + (size_t)N_NODES * H2;          // 50000*256
    float* hW2   = out1 + (size_t)N_NODES * H2;          // 50000*128
    float* z     = hW2  + (size_t)N_NODES * H1;          // 50000*128
    float* meanz = z    + (size_t)N_NODES * H1;          // 128

    // 1) degrees (with self loops) -> dinv; node mask for layer-2 pruning
    k_init  <<<(N_NODES + 255) / 256, 256, 0, stream>>>(dinv, mask);
    k_mask  <<<(N_POS   + 255) / 256, 256, 0, stream>>>(xpos, mask);
    k_deg   <<<(N_EDGES + 255) / 256, 256, 0, stream>>>(ecol, dinv);
    k_rsqrt <<<(N_NODES + 255) / 256, 256, 0, stream>>>(dinv);

    // 2) layer 1 lin: hW1 = emb[x] @ W1   (WMMA fp32, gather via x)
    k_gemm_wmma<1, 0, EMB_DIM, H2>
        <<<dim3(N_NODES / 16, H2 / 128), 256, 0, stream>>>(emb, x, W1, hW1);

    // 3) layer 1 aggregate: out1 = selfloop + b1, then edge scatter
    k_self_bias<<<(N_NODES * H2) / 256, 256, 0, stream>>>(
        hW1, out1, dinv, b1, (unsigned)N_NODES * H2, /*dimShift=*/8);
    k_edge_scatter<<<((unsigned)N_EDGES * H2) / 256, 256, 0, stream>>>(
        hW1, out1, dinv, erow, ecol, /*mask=*/nullptr, /*dimShift=*/8);

    // 4) layer 2 lin: hW2 = relu(out1) @ W2   (relu fused into A-fragment read)
    k_gemm_wmma<0, 1, H2, H1>
        <<<dim3(N_NODES / 16, 1), 256, 0, stream>>>(out1, nullptr, W2, hW2);

    // 5) layer 2 aggregate (only destinations in x_position matter)
    k_self_bias<<<(N_NODES * H1) / 256, 256, 0, stream>>>(
        hW2, z, dinv, b2, (unsigned)N_NODES * H1, /*dimShift=*/7);
    k_edge_scatter<<<((unsigned)N_EDGES * H1) / 256, 256, 0, stream>>>(
        hW2, z, dinv, erow, ecol, mask, /*dimShift=*/7);

    // 6) out = mean(z[x_position]) @ Wc + bc
    k_mean_pos<<<1, H1, 0, stream>>>(z, xpos, meanz);
    k_matvec  <<<VOCAB / 256, 256, 0, stream>>>(meanz, Wc, bc, out);
}